// Lineq2v2_42709154791756
// MI455X (gfx1250) — compile-verified
//
#include <hip/hip_runtime.h>
#include <hip/hip_bf16.h>

typedef __attribute__((ext_vector_type(2))) float v2f;
typedef __attribute__((ext_vector_type(8))) float v8f;

#define NN 128
#define LL 16
#define FF 32
#define INV_NAVG (1.0f / 50.0f)

// ---------------- Kernel 1: rowsum[b,i,l] = (1/NAVG) * sum_j in[b,i,j,l] ----------------
__global__ void __launch_bounds__(128) lineq_rowsum(const float* __restrict__ in,
                                                    float* __restrict__ rowsum) {
    __shared__ float s[128];
    const int bi = blockIdx.x;            // b*N + i
    const int tid = threadIdx.x;
    const int l = tid & 15;
    const int js = tid >> 4;              // 8 stripes of 16 j's
    const float* p = in + ((size_t)bi * NN + js * 16) * LL + l;
    float acc = 0.0f;
#pragma unroll
    for (int j = 0; j < 16; ++j) acc += p[(size_t)j * LL];
    s[tid] = acc;
    __syncthreads();
    if (tid < 16) {
        float t = 0.0f;
#pragma unroll
        for (int g = 0; g < 8; ++g) t += s[g * 16 + tid];
        rowsum[(size_t)bi * LL + tid] = t * INV_NAVG;
    }
}

// ---------------- Kernel 1b: totsum[b,l] = (1/NAVG) * sum_i rowsum[b,i,l] ----------------
__global__ void __launch_bounds__(128) lineq_totsum(const float* __restrict__ rowsum,
                                                    float* __restrict__ totsum) {
    __shared__ float s[128];
    const int b = blockIdx.x;
    const int tid = threadIdx.x;
    const int l = tid & 15;
    const int is = tid >> 4;
    float acc = 0.0f;
    for (int i = is; i < NN; i += 8) acc += rowsum[((size_t)b * NN + i) * LL + l];
    s[tid] = acc;
    __syncthreads();
    if (tid < 16) {
        float t = 0.0f;
#pragma unroll
        for (int g = 0; g < 8; ++g) t += s[g * 16 + tid];
        totsum[b * LL + tid] = t * INV_NAVG;
    }
}

// ---------------- Kernel 2: fold all rank-1 / broadcast terms into 3 small tables ------
// R [b,n,f] = bias[f] + totsum[b]·w1[:,f] + rowsum[b,n]·w2[:,f]   (added per output row i)
// C3[b,n,f] =                               rowsum[b,n]·w3[:,f]   (added per output col j)
// Dg[b,n,f] = diag_bias[f] + totsum[b]·w5[:,f] + rowsum[b,n]·w4[:,f]  (added when i==j)
__global__ void __launch_bounds__(256) lineq_prep(const float* __restrict__ rowsum,
                                                  const float* __restrict__ totsum,
                                                  const float* __restrict__ w,
                                                  const float* __restrict__ bias,
                                                  const float* __restrict__ diag_bias,
                                                  float* __restrict__ R,
                                                  float* __restrict__ C3,
                                                  float* __restrict__ Dg) {
    const int t = blockIdx.x * 256 + threadIdx.x;   // B*N*F threads
    const int f = t & (FF - 1);
    const int bn = t >> 5;                          // b*N + n
    const int b = bn >> 7;
    const float* rs = rowsum + (size_t)bn * LL;
    const float* ts = totsum + (size_t)b * LL;
    float t1 = 0.f, r2 = 0.f, r3 = 0.f, r4 = 0.f, t5 = 0.f;
#pragma unroll
    for (int l = 0; l < LL; ++l) {
        const float rv = rs[l], tv = ts[l];
        const float* wl = w + (size_t)l * 6 * FF;   // w[l, op, f]
        t1 += tv * wl[1 * FF + f];
        r2 += rv * wl[2 * FF + f];
        r3 += rv * wl[3 * FF + f];
        r4 += rv * wl[4 * FF + f];
        t5 += tv * wl[5 * FF + f];
    }
    R [(size_t)bn * FF + f] = bias[f] + t1 + r2;
    C3[(size_t)bn * FF + f] = r3;
    Dg[(size_t)bn * FF + f] = diag_bias[f] + t5 + r4;
}

// ---------------- Kernel 3: main WMMA pass ----------------
// Block = one (b,i) row (8192 blocks), 8 waves, wave w = j-tile [16w, 16w+16).
// Per wave: A = in[b,i, j0:j0+16, 0:16] (16x16 fp32, 4 K-chunks of 4),
//           B = w0 columns (two 16x16 tiles, F 0-15 / 16-31),
//           C seeded with R + C3 (+Dg on diagonal) -> chained v_wmma_f32_16x16x4_f32,
//           ReLU, store 16x32 fp32 tile.
__global__ void __launch_bounds__(256) lineq_main(const float* __restrict__ in,
                                                  const float* __restrict__ w,
                                                  const float* __restrict__ R,
                                                  const float* __restrict__ C3,
                                                  const float* __restrict__ Dg,
                                                  float* __restrict__ out) {
    const int bi = blockIdx.x;            // b*N + i
    const int b = bi >> 7;
    const int i = bi & (NN - 1);
    const int wave = threadIdx.x >> 5;
    const int lane = threadIdx.x & 31;
    const int half = lane >> 4;           // 0: lanes 0-15, 1: lanes 16-31
    const int ln = lane & 15;
    const int j0 = wave * 16;

    // --- B fragments: w0[l,f] = w[l*6*FF + f].  32-bit 4x16 B tile layout mirrors A:
    // lane holds column N=ln, rows K = 4c+2*half, 4c+2*half+1.
    v2f bf[2][4];
#pragma unroll
    for (int fh = 0; fh < 2; ++fh) {
        const int f = fh * 16 + ln;
#pragma unroll
        for (int c = 0; c < 4; ++c) {
            const int k0 = 4 * c + 2 * half;
            bf[fh][c].x = w[(size_t)(k0    ) * 6 * FF + f];
            bf[fh][c].y = w[(size_t)(k0 + 1) * 6 * FF + f];
        }
    }

    // --- Seed accumulators with all additive terms.
    // D layout: VGPR r, lane -> (pixel m = r + 8*half, f = ln + 16*fh)
    const size_t rowbase = (size_t)bi * FF;                 // R, Dg at (b,i)
    const size_t colbase = ((size_t)b * NN + j0) * FF;      // C3 at (b, j0+m)
    const bool tile_has_diag = ((i >> 4) == wave);
    v8f acc[2];
#pragma unroll
    for (int fh = 0; fh < 2; ++fh) {
        const int f = fh * 16 + ln;
        const float radd = R[rowbase + f];
        const float dgf = tile_has_diag ? Dg[rowbase + f] : 0.0f;
#pragma unroll
        for (int r = 0; r < 8; ++r) {
            const int m = r + 8 * half;
            float v = radd + C3[colbase + (size_t)m * FF + f];
            v += ((j0 + m) == i) ? dgf : 0.0f;   // cndmask, no EXEC divergence at WMMA
            acc[fh][r] = v;
        }
    }

    // --- A fragments + chained WMMAs.
    // 32-bit A 16x4: lane holds row M=ln, K = 4c + 2*half + {0,1} (two contiguous floats).
    const float* rowp = in + (((size_t)bi) * NN + j0 + ln) * LL + half * 2;
#pragma unroll
    for (int c = 0; c < 4; ++c) {
        const v2f a = *(const v2f*)(rowp + c * 4);
        acc[0] = __builtin_amdgcn_wmma_f32_16x16x4_f32(
            false, a, false, bf[0][c], (short)0, acc[0], false, false);
        acc[1] = __builtin_amdgcn_wmma_f32_16x16x4_f32(
            false, a, false, bf[1][c], (short)0, acc[1], false, false);
    }

    // --- ReLU + store (f-major, 32 floats per pixel).
    float* outp = out + ((size_t)bi * NN + j0) * FF;
#pragma unroll
    for (int fh = 0; fh < 2; ++fh) {
        const int f = fh * 16 + ln;
#pragma unroll
        for (int r = 0; r < 8; ++r) {
            const int m = r + 8 * half;
            const float v = acc[fh][r];
            outp[(size_t)m * FF + f] = v > 0.0f ? v : 0.0f;
        }
    }
}

extern "C" void kernel_launch(void* const* d_in, const int* in_sizes, int n_in,
                              void* d_out, int out_size, void* d_ws, size_t ws_size,
                              hipStream_t stream) {
    const float* in        = (const float*)d_in[0];   // (64,128,128,16)
    const float* w         = (const float*)d_in[1];   // (16,6,32)
    const float* bias      = (const float*)d_in[2];   // (32,)
    const float* diag_bias = (const float*)d_in[3];   // (32,)
    float* out = (float*)d_out;

    float* ws = (float*)d_ws;
    float* rowsum = ws;                  // 64*128*16      = 131072 floats
    float* totsum = ws + 131072;         // 64*16          = 1024
    float* R      = ws + 132096;         // 64*128*32      = 262144
    float* C3     = ws + 394240;         // 262144
    float* Dg     = ws + 656384;         // 262144  (total ~3.7 MB)

    lineq_rowsum<<<64 * NN, 128, 0, stream>>>(in, rowsum);
    lineq_totsum<<<64, 128, 0, stream>>>(rowsum, totsum);
    lineq_prep<<<(64 * NN * FF) / 256, 256, 0, stream>>>(rowsum, totsum, w, bias, diag_bias,
                                                         R, C3, Dg);
    lineq_main<<<64 * NN, 256, 0, stream>>>(in, w, R, C3, Dg, out);
}